// Ver2f_27479200760261
// MI455X (gfx1250) — compile-verified
//
#include <hip/hip_runtime.h>
#include <math.h>

#define B_  2
#define N_  16384
#define D_  512
#define H_  256
#define C_  4
#define K_  8
#define TPB 256
#define TILES_PER_BN (N_ / 16)
#define LDS_PITCH 516          // 516 % 64 == 4 -> conflict-free column access
#define CS_BASE 10             // d_out: [0,8)=Y_prob, [8,10)=Y_hat, then scores

typedef float v2f __attribute__((ext_vector_type(2)));
typedef float v8f __attribute__((ext_vector_type(8)));

__device__ __forceinline__ v8f zero8() {
  v8f z = {0.f, 0.f, 0.f, 0.f, 0.f, 0.f, 0.f, 0.f};
  return z;
}

// D = A(16x4,f32) * B(4x16,f32) + C(16x16,f32), wave32
__device__ __forceinline__ v8f wmma4(v2f a, v2f b, v8f c) {
  return __builtin_amdgcn_wmma_f32_16x16x4_f32(
      false, a, false, b, (short)0, c, false, false);
}

__device__ __forceinline__ float sigmoidf_(float x) {
  return 1.f / (1.f + expf(-x));
}

// L2-normalize the 16 rows of a [16][LDS_PITCH] LDS tile (first D_ cols).
// wave w handles rows 2w (lanes 0-15) and 2w+1 (lanes 16-31).
__device__ __forceinline__ void normalize_rows16(float (*tile)[LDS_PITCH],
                                                 int wave, int lane) {
  const int half = lane >> 4;
  const int c    = lane & 15;
  const int r    = wave * 2 + half;
  float s = 0.f;
  for (int d = c; d < D_; d += 16) { float v = tile[r][d]; s += v * v; }
  s += __shfl_xor(s, 1, 32);
  s += __shfl_xor(s, 2, 32);
  s += __shfl_xor(s, 4, 32);
  s += __shfl_xor(s, 8, 32);
  const float inv = 1.f / fmaxf(sqrtf(s), 1e-12f);
  for (int d = c; d < D_; d += 16) tile[r][d] *= inv;
}

__global__ __launch_bounds__(TPB) void Ver2f_fused_tile_kernel(
    const float* __restrict__ x_s, const float* __restrict__ x_l,
    const float* __restrict__ Wp,  const float* __restrict__ bp,
    const float* __restrict__ desc,
    const float* __restrict__ Wv,  const float* __restrict__ bv,
    const float* __restrict__ Wu,  const float* __restrict__ bu,
    const float* __restrict__ wat, const float* __restrict__ bat,
    float* __restrict__ out, float* __restrict__ wsf) {
  __shared__ float tile[16][LDS_PITCH];
  __shared__ float part[8][16];
  __shared__ float evals[16];

  const int tid  = threadIdx.x;
  const int wave = tid >> 5;
  const int lane = tid & 31;
  const int half = lane >> 4;
  const int col  = lane & 15;
  const int krow = 2 * half;

  const int id   = blockIdx.x;
  const int s    = id / (B_ * TILES_PER_BN);
  const int id2  = id - s * (B_ * TILES_PER_BN);
  const int b    = id2 / TILES_PER_BN;
  const int t    = id2 - b * TILES_PER_BN;
  const int row0 = t * 16;

  const float* xin   = (s == 0) ? x_s : x_l;
  const float* xbase = xin + ((size_t)b * N_ + row0) * D_;

  // ---- stage x tile in LDS (coalesced) and l2-normalize rows
  for (int i = tid; i < 16 * D_; i += TPB) {
    const int r = i >> 9, d = i & (D_ - 1);
    tile[r][d] = xbase[(size_t)r * D_ + d];
  }
  __syncthreads();
  normalize_rows16(tile, wave, lane);
  __syncthreads();

  // ---- GEMM1: xp = l2norm(x) @ W_proj  (each wave: 64 output columns)
  v8f acc0 = zero8(), acc1 = zero8(), acc2 = zero8(), acc3 = zero8();
  for (int k0 = 0; k0 < D_; k0 += 4) {
    v2f a;
    a.x = tile[col][k0 + krow];
    a.y = tile[col][k0 + krow + 1];
    const float* w0 = Wp + (size_t)(k0 + krow) * D_ + wave * 64 + col;
    v2f bf;
    bf.x = w0[0];  bf.y = w0[D_];        acc0 = wmma4(a, bf, acc0);
    bf.x = w0[16]; bf.y = w0[D_ + 16];   acc1 = wmma4(a, bf, acc1);
    bf.x = w0[32]; bf.y = w0[D_ + 32];   acc2 = wmma4(a, bf, acc2);
    bf.x = w0[48]; bf.y = w0[D_ + 48];   acc3 = wmma4(a, bf, acc3);
  }
  __syncthreads();  // everyone done reading A from LDS
  {
    const int n0 = wave * 64 + col;
    const float b0 = bp[n0], b1 = bp[n0 + 16], b2 = bp[n0 + 32],
                b3 = bp[n0 + 48];
#pragma unroll
    for (int r = 0; r < 8; ++r) {
      const int m = half * 8 + r;
      tile[m][n0]      = acc0[r] + b0;
      tile[m][n0 + 16] = acc1[r] + b1;
      tile[m][n0 + 32] = acc2[r] + b2;
      tile[m][n0 + 48] = acc3[r] + b3;
    }
  }
  __syncthreads();
  normalize_rows16(tile, wave, lane);   // xp = l2norm(... + b)
  __syncthreads();

  // ---- class scores: sim = xp @ desc^T  (waves 0,1: 16 (c,k) columns each)
  if (wave < 2) {
    v8f sacc = zero8();
    const float* drow = desc + (size_t)(wave * 16 + col) * D_;
    for (int k0 = 0; k0 < D_; k0 += 4) {
      v2f a;
      a.x = tile[col][k0 + krow];
      a.y = tile[col][k0 + krow + 1];
      v2f bf;
      bf.x = drow[k0 + krow];
      bf.y = drow[k0 + krow + 1];
      sacc = wmma4(a, bf, sacc);
    }
    const float scale = 0.044194173824159216f;  // 512^-0.5
#pragma unroll
    for (int r = 0; r < 8; ++r) {
      const float sraw = sacc[r];
      const float ss = sraw * scale;
      float mx = ss;   // softmax over K=8 -> 8-lane groups
      mx = fmaxf(mx, __shfl_xor(mx, 1, 32));
      mx = fmaxf(mx, __shfl_xor(mx, 2, 32));
      mx = fmaxf(mx, __shfl_xor(mx, 4, 32));
      const float e = expf(ss - mx);
      float Z = e, num = e * sraw;
      Z += __shfl_xor(Z, 1, 32); num += __shfl_xor(num, 1, 32);
      Z += __shfl_xor(Z, 2, 32); num += __shfl_xor(num, 2, 32);
      Z += __shfl_xor(Z, 4, 32); num += __shfl_xor(num, 4, 32);
      if ((lane & 7) == 0) {
        const int c    = wave * 2 + ((lane >> 3) & 1);
        const int mrow = half * 8 + r;
        out[CS_BASE + (size_t)s * ((size_t)B_ * N_ * C_) +
            ((size_t)(b * N_ + row0 + mrow)) * C_ + c] = num / Z;
      }
    }
  }

  // ---- gated attention: V = tanh(xp@Wv+bv), U = sigmoid(xp@Wu+bu)
  {
    const int n0 = wave * 32 + col;  // each wave: 32 H-columns
    v8f v0 = zero8(), v1 = zero8(), u0 = zero8(), u1 = zero8();
    for (int k0 = 0; k0 < D_; k0 += 4) {
      v2f a;
      a.x = tile[col][k0 + krow];
      a.y = tile[col][k0 + krow + 1];
      const float* pv = Wv + (size_t)(k0 + krow) * H_ + n0;
      const float* pu = Wu + (size_t)(k0 + krow) * H_ + n0;
      v2f bf;
      bf.x = pv[0];  bf.y = pv[H_];        v0 = wmma4(a, bf, v0);
      bf.x = pv[16]; bf.y = pv[H_ + 16];   v1 = wmma4(a, bf, v1);
      bf.x = pu[0];  bf.y = pu[H_];        u0 = wmma4(a, bf, u0);
      bf.x = pu[16]; bf.y = pu[H_ + 16];   u1 = wmma4(a, bf, u1);
    }
    const float bv0 = bv[n0],  bv1 = bv[n0 + 16];
    const float bu0 = bu[n0],  bu1 = bu[n0 + 16];
    const float wa0 = wat[n0], wa1 = wat[n0 + 16];
    float p[8];
#pragma unroll
    for (int r = 0; r < 8; ++r) {
      const float g0 = tanhf(v0[r] + bv0) * sigmoidf_(u0[r] + bu0);
      const float g1 = tanhf(v1[r] + bv1) * sigmoidf_(u1[r] + bu1);
      p[r] = g0 * wa0 + g1 * wa1;
      p[r] += __shfl_xor(p[r], 1, 32);
      p[r] += __shfl_xor(p[r], 2, 32);
      p[r] += __shfl_xor(p[r], 4, 32);
      p[r] += __shfl_xor(p[r], 8, 32);
    }
    if (col == 0) {
#pragma unroll
      for (int r = 0; r < 8; ++r) part[wave][half * 8 + r] = p[r];
    }
  }
  __syncthreads();
  if (tid < 16) {  // fixed-order combine across waves -> deterministic
    float A = bat[0];
#pragma unroll
    for (int w = 0; w < 8; ++w) A += part[w][tid];
    evals[tid] = expf(A);  // |A| <~ 13, safe unnormalized
  }
  __syncthreads();

  // ---- per-block partial of  sum_n exp(A_n) * xp_n  -> workspace
  {
    const size_t base = (((size_t)s * B_ + b) * TILES_PER_BN + t) * D_;
    for (int d = tid; d < D_; d += TPB) {
      float ssum = 0.f;
#pragma unroll
      for (int m = 0; m < 16; ++m) ssum += evals[m] * tile[m][d];
      wsf[base + d] = ssum;
    }
  }
}

__device__ __forceinline__ void block_l2norm_512(float* vec, float* red,
                                                 int tid) {
  float ps = 0.f;
  for (int d = tid; d < D_; d += TPB) { float v = vec[d]; ps += v * v; }
  red[tid] = ps;
  __syncthreads();
  for (int off = TPB / 2; off > 0; off >>= 1) {
    if (tid < off) red[tid] += red[tid + off];
    __syncthreads();
  }
  const float inv = 1.f / fmaxf(sqrtf(red[0]), 1e-12f);
  for (int d = tid; d < D_; d += TPB) vec[d] *= inv;
  __syncthreads();
}

__global__ __launch_bounds__(TPB) void Ver2f_finalize_kernel(
    const float* __restrict__ desc, const float* __restrict__ wsf,
    float* __restrict__ out) {
  __shared__ float slide[4][D_];   // [s*B+b][d]
  __shared__ float tf[C_][D_];
  __shared__ float red[TPB];
  __shared__ float logits[B_][C_];
  const int tid = threadIdx.x;

  // deterministic fixed-order reduction of per-tile partials
  for (int sb = 0; sb < 4; ++sb) {
    for (int d = tid; d < D_; d += TPB) {
      float acc = 0.f;
      const float* p = wsf + ((size_t)sb * TILES_PER_BN) * D_ + d;
      for (int t = 0; t < TILES_PER_BN; ++t) acc += p[(size_t)t * D_];
      slide[sb][d] = acc;  // softmax denom cancels under l2norm
    }
  }
  __syncthreads();
  for (int sb = 0; sb < 4; ++sb) block_l2norm_512(slide[sb], red, tid);

  // text feats: l2norm(max_k desc[c,k,:])
  for (int c = 0; c < C_; ++c) {
    for (int d = tid; d < D_; d += TPB) {
      float m = desc[(size_t)(c * K_) * D_ + d];
#pragma unroll
      for (int k = 1; k < K_; ++k)
        m = fmaxf(m, desc[(size_t)(c * K_ + k) * D_ + d]);
      tf[c][d] = m;
    }
  }
  __syncthreads();
  for (int c = 0; c < C_; ++c) block_l2norm_512(tf[c], red, tid);

  if (tid < B_ * C_) {
    const int b = tid / C_, c = tid % C_;
    float acc = 0.f;
    for (int d = 0; d < D_; ++d)
      acc += (slide[b][d] + slide[2 + b][d]) * tf[c][d];
    logits[b][c] = acc;
  }
  __syncthreads();
  if (tid < B_) {
    const int b = tid;
    float m = logits[b][0];
    int am = 0;
    for (int c = 1; c < C_; ++c)
      if (logits[b][c] > m) { m = logits[b][c]; am = c; }
    float e[C_], Z = 0.f;
    for (int c = 0; c < C_; ++c) { e[c] = expf(logits[b][c] - m); Z += e[c]; }
    for (int c = 0; c < C_; ++c) out[b * C_ + c] = e[c] / Z;
    out[B_ * C_ + b] = (float)am;  // Y_hat
  }
}

extern "C" void kernel_launch(void* const* d_in, const int* in_sizes, int n_in,
                              void* d_out, int out_size, void* d_ws,
                              size_t ws_size, hipStream_t stream) {
  (void)in_sizes; (void)n_in; (void)out_size; (void)ws_size;
  const float* x_s  = (const float*)d_in[0];
  const float* x_l  = (const float*)d_in[2];   // coords (1,3) unused
  const float* Wp   = (const float*)d_in[4];
  const float* bp   = (const float*)d_in[5];
  const float* desc = (const float*)d_in[6];
  const float* Wv   = (const float*)d_in[7];
  const float* bv   = (const float*)d_in[8];
  const float* Wu   = (const float*)d_in[9];
  const float* bu   = (const float*)d_in[10];
  const float* wat  = (const float*)d_in[11];
  const float* bat  = (const float*)d_in[12];
  float* out = (float*)d_out;
  float* wsf = (float*)d_ws;  // 2*2*1024*512 floats = 8 MB partials

  dim3 grid(2 * B_ * TILES_PER_BN);  // 4096 tiles of 16 rows
  Ver2f_fused_tile_kernel<<<grid, dim3(TPB), 0, stream>>>(
      x_s, x_l, Wp, bp, desc, Wv, bv, Wu, bu, wat, bat, out, wsf);
  Ver2f_finalize_kernel<<<dim3(1), dim3(TPB), 0, stream>>>(desc, wsf, out);
}